// RoIAlign2D_33423435498476
// MI455X (gfx1250) — compile-verified
//
#include <hip/hip_runtime.h>
#include <hip/hip_bf16.h>
#include <stdint.h>

// RoI Align, NCHW f32, OUT_SIZE=7, SAMPLE_NUM=2, scale=1/16.
// Memory-bound gather: per-roi window staged to LDS with gfx1250 async
// global->LDS loads (ASYNCcnt), bilinear interpolation from LDS.
// WMMA intentionally avoided (dense formulation inflates FLOPs ~50,000x).

#define ROI_OUT   7
#define NSAMP     14          // OUT_SIZE * SAMPLE_NUM
#define CCH       256
#define HH        96
#define WW        96
#define RSCALE    0.0625f
#define TILE_ROWS 24
#define TILE_W    32
#define NWAVES    8
#define CSPLIT    4           // gridDim.y channel partitions
#define CPB       (CCH / CSPLIT)   // 64 channels per block

__global__ __launch_bounds__(256)
void roialign_gfx1250_kernel(const float* __restrict__ feat,
                             const float* __restrict__ rois,
                             float* __restrict__ out)
{
    __shared__ int   s_ylo[NSAMP], s_yhi[NSAMP], s_xlo[NSAMP], s_xhi[NSAMP];
    __shared__ float s_wy0[NSAMP], s_wy1[NSAMP], s_wx0[NSAMP], s_wx1[NSAMP];
    __shared__ int   s_meta[4];                 // y_base, x_base, rows, batch
    __shared__ float s_tile[NWAVES][TILE_ROWS * TILE_W];

    const int k    = blockIdx.x;
    const int c0   = blockIdx.y * CPB;
    const int tid  = threadIdx.x;
    const int lane = tid & 31;
    const int wave = tid >> 5;

    // ---- per-roi 1-D sample tables (threads 0..27) ----
    if (tid < 2 * NSAMP) {
        const int axis = tid / NSAMP;           // 0 = y, 1 = x
        const int g    = tid % NSAMP;
        const float p1  = rois[k * 5 + (axis ? 1 : 2)] * RSCALE;
        const float p2  = rois[k * 5 + (axis ? 3 : 4)] * RSCALE;
        const float len = fmaxf(p2 - p1, 1.0f);
        const float bin = len * (1.0f / (float)ROI_OUT);
        const float offs  = (float)(g >> 1) + ((float)(g & 1) + 0.5f) * 0.5f;
        const float coord = p1 + bin * offs;
        const int   size  = axis ? WW : HH;
        const bool  valid = (coord >= -1.0f) && (coord <= (float)size);
        const float cc    = fminf(fmaxf(coord, 0.0f), (float)(size - 1));
        const float lo    = floorf(cc);
        const int   ilo   = (int)lo;
        const int   ihi   = min(ilo + 1, size - 1);
        const float fr    = cc - lo;
        const float w0    = valid ? (1.0f - fr) : 0.0f;
        const float w1    = valid ? fr          : 0.0f;
        if (axis == 0) { s_ylo[g] = ilo; s_yhi[g] = ihi; s_wy0[g] = w0; s_wy1[g] = w1; }
        else           { s_xlo[g] = ilo; s_xhi[g] = ihi; s_wx0[g] = w0; s_wx1[g] = w1; }
    }
    if (tid == 0) s_meta[3] = (int)rois[k * 5];
    __syncthreads();

    if (tid == 0) {
        s_meta[0] = s_ylo[0];                                   // window y origin
        s_meta[1] = s_xlo[0];                                   // window x origin
        s_meta[2] = min(s_yhi[NSAMP - 1] - s_ylo[0] + 1, TILE_ROWS);
    }
    __syncthreads();

    // rebase indices to window origin (y premultiplied by tile pitch)
    if (tid < NSAMP) {
        const int yb = s_meta[0];
        s_ylo[tid] = min(s_ylo[tid] - yb, TILE_ROWS - 1) * TILE_W;
        s_yhi[tid] = min(s_yhi[tid] - yb, TILE_ROWS - 1) * TILE_W;
    } else if (tid < 2 * NSAMP) {
        const int g  = tid - NSAMP;
        const int xb = s_meta[1];
        s_xlo[g] = min(s_xlo[g] - xb, TILE_W - 1);
        s_xhi[g] = min(s_xhi[g] - xb, TILE_W - 1);
    }
    __syncthreads();

    const int y_base = s_meta[0];
    const int x_base = s_meta[1];
    const int rows   = s_meta[2];
    const int bidx   = s_meta[3];

    float* tile = &s_tile[wave][0];
    const int  xcol  = min(x_base + lane, WW - 1);  // clamp keeps address in-bounds;
                                                    // clamped lanes feed unread slots
    const long plane = (long)(bidx * CCH) * HH;
    const long row0  = (plane + y_base) * WW + xcol;
    // prefetch base: lane r touches row r of a channel's window
    const long pref0 = (plane + y_base + min(lane, rows - 1)) * WW + x_base;

    for (int c = c0 + wave; c < c0 + CPB; c += NWAVES) {
        // Drain this wave's outstanding LDS reads before async writes can land
        asm volatile("s_wait_dscnt 0x0" ::: "memory");

        // ---- stage window rows: coalesced 128B async global->LDS per row ----
        const float* gp  = feat + row0 + (long)c * (HH * WW);
        uint32_t     lds = (uint32_t)(uintptr_t)(tile + lane);
        for (int r = 0; r < rows; ++r) {
            asm volatile("global_load_async_to_lds_b32 %0, %1, off"
                         :: "v"(lds), "v"(gp) : "memory");
            lds += TILE_W * 4;
            gp  += WW;
        }
        // warm next channel's window in cache (lane-spread over rows)
        const int cn = min(c + NWAVES, CCH - 1);
        __builtin_prefetch(feat + pref0 + (long)cn * (HH * WW), 0, 1);

        asm volatile("s_wait_asynccnt 0x0" ::: "memory");

        // ---- 49 bins across lanes: 2x2 samples, bilinear from LDS ----
        for (int bin = lane; bin < ROI_OUT * ROI_OUT; bin += 32) {
            const int ph = bin / ROI_OUT;
            const int pw = bin - ph * ROI_OUT;
            float acc = 0.0f;
#pragma unroll
            for (int sy = 0; sy < 2; ++sy) {
                const int   i   = 2 * ph + sy;
                const int   y0  = s_ylo[i], y1 = s_yhi[i];
                const float wy0 = s_wy0[i], wy1 = s_wy1[i];
#pragma unroll
                for (int sx = 0; sx < 2; ++sx) {
                    const int   j   = 2 * pw + sx;
                    const int   x0  = s_xlo[j], x1 = s_xhi[j];
                    const float wx0 = s_wx0[j], wx1 = s_wx1[j];
                    const float top = tile[y0 + x0] * wx0 + tile[y0 + x1] * wx1;
                    const float bot = tile[y1 + x0] * wx0 + tile[y1 + x1] * wx1;
                    acc += wy0 * top + wy1 * bot;
                }
            }
            out[((long)(k * CCH + c)) * (ROI_OUT * ROI_OUT) + bin] = acc * 0.25f;
        }
    }
}

extern "C" void kernel_launch(void* const* d_in, const int* in_sizes, int n_in,
                              void* d_out, int out_size, void* d_ws, size_t ws_size,
                              hipStream_t stream) {
    const float* feat = (const float*)d_in[0];   // (4,256,96,96) f32
    const float* rois = (const float*)d_in[1];   // (K,5) f32
    float*       out  = (float*)d_out;           // (K,256,7,7) f32
    const int K = in_sizes[1] / 5;               // 1024
    roialign_gfx1250_kernel<<<dim3(K, CSPLIT), dim3(256), 0, stream>>>(feat, rois, out);
}